// MMSE_Transciever_43765716746670
// MI455X (gfx1250) — compile-verified
//
#include <hip/hip_runtime.h>
#include <math.h>
#include <stdint.h>

#define B_    1024
#define K_    8
#define NRX_  16
#define NTX_  64
#define D_    4
#define ITERS_ 10
#define SIGMA_ 0.01f
#define NEWTON_THRESH_ 1e-7f
#define NEWTON_MAX_ 50

typedef float v2f __attribute__((ext_vector_type(2)));
typedef float v8f __attribute__((ext_vector_type(8)));

// ---------------- WMMA wrapper: V_WMMA_F32_16X16X4_F32 ----------------
// F32 WMMA has no A/B negation modifiers (NEG = CNeg,0,0 per ISA); negate
// fragments in VALU where the complex arithmetic needs "-A*B + C".
__device__ __forceinline__ v8f wmma4(v2f a, v2f b, v8f c) {
  return __builtin_amdgcn_wmma_f32_16x16x4_f32(false, a, false, b, (short)0, c, false, false);
}
__device__ __forceinline__ v2f negv(v2f a) {
  v2f r; r.x = -a.x; r.y = -a.y; return r;
}

// ---------------- CDNA5 async global->LDS copy (ASYNCcnt path) ----------------
// 16B per lane, GLOBAL_LOAD_ASYNC_TO_LDS_B128 (GV mode). LDS operand is the
// low 32 bits of the generic pointer (flat LDS aperture keeps offset in [31:0]).
__device__ __forceinline__ uint32_t lds_off(const void* p) {
  return (uint32_t)(uintptr_t)p;
}
__device__ __forceinline__ void async_copy16(uint32_t lds_addr, const void* gptr) {
  asm volatile("global_load_async_to_lds_b128 %0, %1, off"
               :: "v"(lds_addr), "v"(gptr) : "memory");
}
__device__ __forceinline__ void wait_async() {
  asm volatile("s_wait_asynccnt 0x0" ::: "memory");
}

// A-fragment (16x4 f32) from planar [rows][4] array, rows r0..r0+15.
// ISA layout: lanes 0-15 hold K=0,1 ; lanes 16-31 hold K=2,3.
__device__ __forceinline__ v2f fragA4(const float* M, int r0, int lane) {
  int l = lane & 15, h = (lane >> 4) & 1;
  const float* p = M + (r0 + l) * 4 + 2 * h;
  v2f a; a.x = p[0]; a.y = p[1];
  return a;
}

// gram64: out(64x64 complex) = sum_k Y_k Y_k^H, Y_k (64x4 complex),
// Y stored planar in LDS as P/Q[k][64][4]. 8 waves, 16 output tiles of 16x16.
__device__ void gram64(const float* P, const float* Q,
                       float* __restrict__ gre, float* __restrict__ gim, int tid) {
  int wave = tid >> 5, lane = tid & 31;
  int l = lane & 15, h = (lane >> 4) & 1;
  for (int tile = wave; tile < 16; tile += 8) {
    int ti = tile >> 2, tj = tile & 3;
    v8f cre = {}; v8f cim = {};
    for (int k = 0; k < K_; ++k) {
      const float* Pk = P + k * 256;
      const float* Qk = Q + k * 256;
      v2f aP = fragA4(Pk, ti * 16, lane);
      v2f aQ = fragA4(Qk, ti * 16, lane);
      v2f bP = fragA4(Pk, tj * 16, lane);   // B[kk][n] = P[tj*16+n][kk]
      v2f bQ = fragA4(Qk, tj * 16, lane);
      // Re = P P^T + Q Q^T ; Im = Q P^T - P Q^T
      cre = wmma4(aP, bP, cre);
      cre = wmma4(aQ, bQ, cre);
      cim = wmma4(aQ, bP, cim);
      cim = wmma4(negv(aP), bQ, cim);
    }
    for (int r = 0; r < 8; ++r) {
      int row = ti * 16 + r + h * 8;
      int col = tj * 16 + l;
      gre[row * 64 + col] = cre[r];
      gim[row * 64 + col] = cim[r];
    }
  }
}

// ---------------- Kernel 1: xhu = X^H u ; gram = sum_k xhu xhu^H ----------------
__global__ __launch_bounds__(256) void k_xhu_gram(
    const float* __restrict__ x, const float* __restrict__ uRe, const float* __restrict__ uIm,
    float* __restrict__ xhuRe, float* __restrict__ xhuIm,
    float* __restrict__ gre, float* __restrict__ gim) {
  __shared__ __align__(16) float P[K_ * 256];
  __shared__ __align__(16) float Q[K_ * 256];
  int b = blockIdx.x, tid = threadIdx.x;
  const float* xb = x + (size_t)b * K_ * NRX_ * NTX_ * 2;
  __builtin_prefetch(xb, 0, 1);
  for (int idx = tid; idx < K_ * NTX_; idx += 256) {
    int k = idx >> 6, t = idx & 63;
    const float* xk = xb + k * NRX_ * NTX_ * 2;
    const float* ur = uRe + (size_t)b * 512 + k * 64;
    const float* ui = uIm + (size_t)b * 512 + k * 64;
    float ar[D_] = {0.f, 0.f, 0.f, 0.f}, ai[D_] = {0.f, 0.f, 0.f, 0.f};
    for (int r = 0; r < NRX_; ++r) {
      float xr = xk[(r * NTX_ + t) * 2 + 0];
      float xi = xk[(r * NTX_ + t) * 2 + 1];
      for (int d = 0; d < D_; ++d) {
        float urv = ur[r * 4 + d], uiv = ui[r * 4 + d];
        ar[d] += xr * urv + xi * uiv;     // conj(x)*u
        ai[d] += xr * uiv - xi * urv;
      }
    }
    for (int d = 0; d < D_; ++d) {
      P[k * 256 + t * 4 + d] = ar[d];
      Q[k * 256 + t * 4 + d] = ai[d];
      xhuRe[(size_t)b * 2048 + k * 256 + t * 4 + d] = ar[d];
      xhuIm[(size_t)b * 2048 + k * 256 + t * 4 + d] = ai[d];
    }
  }
  __syncthreads();
  gram64(P, Q, gre + (size_t)b * 4096, gim + (size_t)b * 4096, tid);
}

// ---------------- Kernel 2: v = (gram + lam I)^-1 xhu  (complex Gauss-Jordan) ----------------
__global__ __launch_bounds__(256) void k_solve_v(
    const float* __restrict__ gre, const float* __restrict__ gim,
    const float* __restrict__ lam,
    const float* __restrict__ xhuRe, const float* __restrict__ xhuIm,
    float* __restrict__ vRe, float* __restrict__ vIm) {
  __shared__ __align__(16) float Ar[64 * 96];
  __shared__ __align__(16) float Ai[64 * 96];
  int b = blockIdx.x, tid = threadIdx.x;
  float l = lam[0];
  const float* gr = gre + (size_t)b * 4096;
  const float* gi = gim + (size_t)b * 4096;
  const float* xr = xhuRe + (size_t)b * 2048;
  const float* xi = xhuIm + (size_t)b * 2048;
  // Async-stage gram (row stride 96 in LDS) and the 32 RHS columns, 16B chunks.
  for (int i = tid; i < 2048; i += 256) {
    int comp = i >> 10;                 // 0 = re, 1 = im
    int j = i & 1023;                   // gram chunk: row r, 4-col group
    int r = j >> 4, c4 = (j & 15) * 4;
    const float* src = (comp ? gi : gr) + r * 64 + c4;
    float* dst = (comp ? Ai : Ar) + r * 96 + c4;
    async_copy16(lds_off(dst), src);
  }
  for (int i = tid; i < 1024; i += 256) {
    int comp = i >> 9;
    int j = i & 511;                    // j = k*64 + t
    int k = j >> 6, t = j & 63;
    const float* src = (comp ? xi : xr) + k * 256 + t * 4;
    float* dst = (comp ? Ai : Ar) + t * 96 + 64 + k * 4;
    async_copy16(lds_off(dst), src);
  }
  wait_async();
  __syncthreads();
  if (tid < 64) Ar[tid * 96 + tid] += l;   // ridge diagonal
  __syncthreads();
  for (int p = 0; p < 64; ++p) {
    float pr = Ar[p * 96 + p], pi_ = Ai[p * 96 + p];
    float inv = 1.f / (pr * pr + pi_ * pi_ + 1e-30f);
    float ir = pr * inv, ii = -pi_ * inv;
    for (int c = p + 1 + tid; c < 96; c += 256) {
      float vr = Ar[p * 96 + c], vi = Ai[p * 96 + c];
      Ar[p * 96 + c] = vr * ir - vi * ii;
      Ai[p * 96 + c] = vr * ii + vi * ir;
    }
    __syncthreads();
    int width = 95 - p;
    for (int it2 = tid; it2 < 64 * width; it2 += 256) {
      int r = it2 / width;
      int c = p + 1 + it2 % width;
      if (r == p) continue;
      float fr = Ar[r * 96 + p], fi = Ai[r * 96 + p];
      float vr = Ar[p * 96 + c], vi = Ai[p * 96 + c];
      Ar[r * 96 + c] -= fr * vr - fi * vi;
      Ai[r * 96 + c] -= fr * vi + fi * vr;
    }
    __syncthreads();
  }
  for (int i = tid; i < 2048; i += 256) {
    int k = i >> 8, rem = i & 255, t = rem >> 2, d = rem & 3;
    vRe[(size_t)b * 2048 + i] = Ar[t * 96 + 64 + k * 4 + d];
    vIm[(size_t)b * 2048 + i] = Ai[t * 96 + 64 + k * 4 + d];
  }
}

// ---------------- Kernel 3: Z = X_k [v_0..v_7]; u_sum = Z Z^H + sI; u = u_sum^-1 Z[:,k] ----------------
__global__ __launch_bounds__(256) void k_update_u(
    const float* __restrict__ x,
    const float* __restrict__ vRe, const float* __restrict__ vIm,
    float* __restrict__ uRe, float* __restrict__ uIm) {
  __shared__ __align__(16) float smem[12288];
  float* Pv = smem;          // 2048: v planar [k][64][4] (re)
  float* Qv = smem + 2048;   // 2048: (im)
  float* Zp = smem + 4096;   // 4096: Z [k][16][32] (re)
  float* Zq = smem + 8192;   // 4096: (im)
  int b = blockIdx.x, tid = threadIdx.x;
  int k = tid >> 5, lane = tid & 31;
  int l = lane & 15, h = (lane >> 4) & 1;
  // Async-stage the whole per-batch v (2 x 8KB) into LDS.
  for (int i = tid; i < 1024; i += 256) {
    int comp = i >> 9;
    int j = (i & 511) * 4;
    const float* src = (comp ? vIm : vRe) + (size_t)b * 2048 + j;
    float* dst = (comp ? Qv : Pv) + j;
    async_copy16(lds_off(dst), src);
  }
  wait_async();
  __syncthreads();
  const float* xk = x + ((size_t)b * K_ + k) * NRX_ * NTX_ * 2;
  __builtin_prefetch(xk, 0, 1);
  // Z = X_k @ Vc (16x32 complex), inner dim 64 in 16 WMMA k-chunks, 2 N-tiles
  v8f zr0 = {}, zi0 = {}, zr1 = {}, zi1 = {};
  for (int c = 0; c < 16; ++c) {
    int col0 = 4 * c + 2 * h;
    const float* xrow = xk + (l * NTX_ + col0) * 2;
    v2f aR, aI;
    aR.x = xrow[0]; aI.x = xrow[1];
    aR.y = xrow[2]; aI.y = xrow[3];
    v2f aIn = negv(aI);
    #pragma unroll
    for (int tj = 0; tj < 2; ++tj) {
      int coln = tj * 16 + l;
      int kj = coln >> 2, dj = coln & 3;
      v2f bR, bI;
      bR.x = Pv[kj * 256 + (col0 + 0) * 4 + dj];
      bR.y = Pv[kj * 256 + (col0 + 1) * 4 + dj];
      bI.x = Qv[kj * 256 + (col0 + 0) * 4 + dj];
      bI.y = Qv[kj * 256 + (col0 + 1) * 4 + dj];
      if (tj == 0) {
        zr0 = wmma4(aR, bR, zr0); zr0 = wmma4(aIn, bI, zr0);
        zi0 = wmma4(aR, bI, zi0); zi0 = wmma4(aI, bR, zi0);
      } else {
        zr1 = wmma4(aR, bR, zr1); zr1 = wmma4(aIn, bI, zr1);
        zi1 = wmma4(aR, bI, zi1); zi1 = wmma4(aI, bR, zi1);
      }
    }
  }
  float* zpk = Zp + k * 512;
  float* zqk = Zq + k * 512;
  for (int r = 0; r < 8; ++r) {
    int row = r + h * 8;
    zpk[row * 32 + l]      = zr0[r];
    zqk[row * 32 + l]      = zi0[r];
    zpk[row * 32 + 16 + l] = zr1[r];
    zqk[row * 32 + 16 + l] = zi1[r];
  }
  __syncthreads();
  // u_sum = Z Z^H (16x16), inner dim 32 in 8 WMMA chunks; A and B frags share lane addressing
  v8f sr = {}, si = {};
  for (int c = 0; c < 8; ++c) {
    int kk0 = 4 * c + 2 * h;
    v2f aR, aI;
    aR.x = zpk[l * 32 + kk0]; aR.y = zpk[l * 32 + kk0 + 1];
    aI.x = zqk[l * 32 + kk0]; aI.y = zqk[l * 32 + kk0 + 1];
    sr = wmma4(aR, aR, sr);
    sr = wmma4(aI, aI, sr);
    si = wmma4(aI, aR, si);
    si = wmma4(negv(aR), aI, si);
  }
  // rhs = X_k v_k = Z[:, k*4..k*4+3] -> keep in regs across the overlay barrier
  float rr[2], ri[2];
  for (int t2 = 0; t2 < 2; ++t2) {
    int item = lane * 2 + t2;       // 0..63
    int r = item >> 2, d = item & 3;
    rr[t2] = zpk[r * 32 + k * 4 + d];
    ri[t2] = zqk[r * 32 + k * 4 + d];
  }
  __syncthreads();
  // overlay augmented systems [8][16][20] onto smem (Pv/Qv/Z dead now)
  float* augR = smem;          // 2560
  float* augI = smem + 2560;   // 2560
  float* aRk = augR + k * 320;
  float* aIk = augI + k * 320;
  for (int r = 0; r < 8; ++r) {
    int row = r + h * 8;
    aRk[row * 20 + l] = sr[r] + ((row == l) ? SIGMA_ : 0.f);
    aIk[row * 20 + l] = si[r];
  }
  for (int t2 = 0; t2 < 2; ++t2) {
    int item = lane * 2 + t2;
    int r = item >> 2, d = item & 3;
    aRk[r * 20 + 16 + d] = rr[t2];
    aIk[r * 20 + 16 + d] = ri[t2];
  }
  __syncthreads();
  // per-wave complex Gauss-Jordan 16x(16+4)
  for (int p = 0; p < 16; ++p) {
    float pr = aRk[p * 20 + p], pi_ = aIk[p * 20 + p];
    float inv = 1.f / (pr * pr + pi_ * pi_ + 1e-30f);
    float ir = pr * inv, ii = -pi_ * inv;
    for (int cc = p + 1 + lane; cc < 20; cc += 32) {
      float vr = aRk[p * 20 + cc], vi = aIk[p * 20 + cc];
      aRk[p * 20 + cc] = vr * ir - vi * ii;
      aIk[p * 20 + cc] = vr * ii + vi * ir;
    }
    __syncthreads();
    int width = 19 - p;
    for (int it2 = lane; it2 < 16 * width; it2 += 32) {
      int r = it2 / width;
      int cc = p + 1 + it2 % width;
      if (r == p) continue;
      float fr = aRk[r * 20 + p], fi = aIk[r * 20 + p];
      float vr = aRk[p * 20 + cc], vi = aIk[p * 20 + cc];
      aRk[r * 20 + cc] -= fr * vr - fi * vi;
      aIk[r * 20 + cc] -= fr * vi + fi * vr;
    }
    __syncthreads();
  }
  for (int t2 = lane; t2 < 64; t2 += 32) {
    int r = t2 >> 2, d = t2 & 3;
    uRe[(size_t)b * 512 + k * 64 + t2] = aRk[r * 20 + 16 + d];
    uIm[(size_t)b * 512 + k * 64 + t2] = aIk[r * 20 + 16 + d];
  }
}

// ---------------- Kernel 4: diag_eig[b] = b-th singular value of gram2[b] ----------------
// gram2 = Y Y^H has rank <= 32 (Y = [xhu2_k] 64x32), so eig(gram2) = eig(Y^H Y) ++ zeros.
__global__ __launch_bounds__(64) void k_eig(
    const float* __restrict__ x, const float* __restrict__ uRe, const float* __restrict__ uIm,
    float* __restrict__ diag_eig) {
  int b = blockIdx.x, tid = threadIdx.x;
  if (b >= 32) { if (tid == 0) diag_eig[b] = 0.f; return; }
  __shared__ float Yr[2048], Yi[2048];   // [k][64][4]
  __shared__ float Sr[1024], Si[1024];   // 32x32
  const float* xb = x + (size_t)b * K_ * NRX_ * NTX_ * 2;
  for (int idx = tid; idx < K_ * NTX_; idx += 64) {
    int k = idx >> 6, t = idx & 63;
    const float* xk = xb + k * NRX_ * NTX_ * 2;
    const float* ur = uRe + (size_t)b * 512 + k * 64;
    const float* ui = uIm + (size_t)b * 512 + k * 64;
    float ar[D_] = {0.f, 0.f, 0.f, 0.f}, ai[D_] = {0.f, 0.f, 0.f, 0.f};
    for (int r = 0; r < NRX_; ++r) {
      float xr = xk[(r * NTX_ + t) * 2 + 0];
      float xi = xk[(r * NTX_ + t) * 2 + 1];
      for (int d = 0; d < D_; ++d) {
        float urv = ur[r * 4 + d], uiv = ui[r * 4 + d];
        ar[d] += xr * urv + xi * uiv;
        ai[d] += xr * uiv - xi * urv;
      }
    }
    for (int d = 0; d < D_; ++d) {
      Yr[k * 256 + t * 4 + d] = ar[d];
      Yi[k * 256 + t * 4 + d] = ai[d];
    }
  }
  __syncthreads();
  for (int i = tid; i < 1024; i += 64) {
    int a = i >> 5, c = i & 31;
    int ka = a >> 2, da = a & 3, kc = c >> 2, dc = c & 3;
    float sr = 0.f, si = 0.f;
    for (int t = 0; t < 64; ++t) {
      float arv = Yr[ka * 256 + t * 4 + da], aiv = Yi[ka * 256 + t * 4 + da];
      float crv = Yr[kc * 256 + t * 4 + dc], civ = Yi[kc * 256 + t * 4 + dc];
      sr += arv * crv + aiv * civ;       // conj(col_a) . col_c
      si += arv * civ - aiv * crv;
    }
    Sr[i] = sr; Si[i] = si;
  }
  __syncthreads();
  if (tid == 0) {
    for (int sweep = 0; sweep < 12; ++sweep) {
      for (int p = 0; p < 31; ++p) {
        for (int q = p + 1; q < 32; ++q) {
          float mr = Sr[p * 32 + q], mi = Si[p * 32 + q];
          float m2 = mr * mr + mi * mi;
          if (m2 < 1e-18f) continue;
          float m = sqrtf(m2);
          float er = mr / m, ei = mi / m;            // e^{i phi}
          float app = Sr[p * 32 + p], aqq = Sr[q * 32 + q];
          float theta = 0.5f * atan2f(2.f * m, app - aqq);
          float cc = cosf(theta), ss = sinf(theta);
          for (int r = 0; r < 32; ++r) {             // column rotation
            float vpr = Sr[r * 32 + p], vpi = Si[r * 32 + p];
            float vqr = Sr[r * 32 + q], vqi = Si[r * 32 + q];
            float eqr = er * vqr - ei * vqi, eqi = er * vqi + ei * vqr;
            Sr[r * 32 + p] = cc * vpr - ss * eqr; Si[r * 32 + p] = cc * vpi - ss * eqi;
            float epr = er * vpr + ei * vpi, epi = er * vpi - ei * vpr;
            Sr[r * 32 + q] = ss * epr + cc * vqr; Si[r * 32 + q] = ss * epi + cc * vqi;
          }
          for (int c2 = 0; c2 < 32; ++c2) {          // row rotation (conjugate)
            float vpr = Sr[p * 32 + c2], vpi = Si[p * 32 + c2];
            float vqr = Sr[q * 32 + c2], vqi = Si[q * 32 + c2];
            float eqr = er * vqr + ei * vqi, eqi = er * vqi - ei * vqr;
            Sr[p * 32 + c2] = cc * vpr - ss * eqr; Si[p * 32 + c2] = cc * vpi - ss * eqi;
            float epr = er * vpr - ei * vpi, epi = er * vpi + ei * vpr;
            Sr[q * 32 + c2] = ss * epr + cc * vqr; Si[q * 32 + c2] = ss * epi + cc * vqi;
          }
        }
      }
    }
    float ev[32];
    for (int i2 = 0; i2 < 32; ++i2) ev[i2] = Sr[i2 * 32 + i2];
    for (int i2 = 1; i2 < 32; ++i2) {                // insertion sort descending
      float key = ev[i2]; int j = i2 - 1;
      while (j >= 0 && ev[j] < key) { ev[j + 1] = ev[j]; --j; }
      ev[j + 1] = key;
    }
    diag_eig[b] = ev[b];
  }
}

// ---------------- Kernel 5: scalar Newton on lam ----------------
__global__ void k_newton(const float* __restrict__ diag_eig, float* __restrict__ lam) {
  if (threadIdx.x != 0 || blockIdx.x != 0) return;
  float l = *lam;
  for (int it = 0; it < NEWTON_MAX_; ++it) {
    float f = -1.f, df = 0.f;
    for (int i = 0; i < 64; ++i) {
      float s = diag_eig[i];
      float den = s + l;
      f += s / den;
      df -= s / (den * den);
    }
    if (fabsf(f) > NEWTON_THRESH_) l = l - f / (df - 1e-30f);
  }
  *lam = l;
}

// ---------------- init / writeout ----------------
__global__ void k_init(const float* __restrict__ v0, const float* __restrict__ u0,
                       float* uRe, float* uIm, float* vRe, float* vIm, float* lam) {
  size_t i = (size_t)blockIdx.x * 256 + threadIdx.x;
  const size_t nu = (size_t)B_ * K_ * NRX_ * D_;
  const size_t nv = (size_t)B_ * K_ * NTX_ * D_;
  if (i < nu) { uRe[i] = u0[2 * i]; uIm[i] = u0[2 * i + 1]; }
  if (i < nv) { vRe[i] = v0[2 * i]; vIm[i] = v0[2 * i + 1]; }
  if (i == 0) *lam = 1.0f;
}

__global__ void k_out(const float* __restrict__ uRe, const float* __restrict__ uIm,
                      const float* __restrict__ vRe, const float* __restrict__ vIm,
                      float* __restrict__ out) {
  size_t i = (size_t)blockIdx.x * 256 + threadIdx.x;
  const size_t nu = (size_t)B_ * K_ * NRX_ * D_;
  const size_t nv = (size_t)B_ * K_ * NTX_ * D_;
  if (i < nu) { out[2 * i] = uRe[i]; out[2 * i + 1] = uIm[i]; }
  if (i < nv) { out[2 * nu + 2 * i] = vRe[i]; out[2 * nu + 2 * i + 1] = vIm[i]; }
}

extern "C" void kernel_launch(void* const* d_in, const int* in_sizes, int n_in,
                              void* d_out, int out_size, void* d_ws, size_t ws_size,
                              hipStream_t stream) {
  (void)in_sizes; (void)n_in; (void)out_size; (void)ws_size;
  const float* x  = (const float*)d_in[0];
  const float* v0 = (const float*)d_in[1];
  const float* u0 = (const float*)d_in[2];
  float* out = (float*)d_out;

  const size_t nu = (size_t)B_ * K_ * NRX_ * D_;   //   524288
  const size_t nv = (size_t)B_ * K_ * NTX_ * D_;   //  2097152
  const size_t ng = (size_t)B_ * 64 * 64;          //  4194304
  float* p = (float*)d_ws;
  float* uRe = p;   p += nu;
  float* uIm = p;   p += nu;
  float* vRe = p;   p += nv;
  float* vIm = p;   p += nv;
  float* hRe = p;   p += nv;   // xhu
  float* hIm = p;   p += nv;
  float* gre = p;   p += ng;   // gram
  float* gim = p;   p += ng;
  float* diag = p;  p += 64;
  float* lam = p;   p += 1;

  k_init<<<8192, 256, 0, stream>>>(v0, u0, uRe, uIm, vRe, vIm, lam);
  for (int it = 0; it < ITERS_; ++it) {
    k_xhu_gram<<<B_, 256, 0, stream>>>(x, uRe, uIm, hRe, hIm, gre, gim);
    k_solve_v<<<B_, 256, 0, stream>>>(gre, gim, lam, hRe, hIm, vRe, vIm);
    k_update_u<<<B_, 256, 0, stream>>>(x, vRe, vIm, uRe, uIm);
    k_eig<<<64, 64, 0, stream>>>(x, uRe, uIm, diag);
    k_newton<<<1, 32, 0, stream>>>(diag, lam);
  }
  k_out<<<8192, 256, 0, stream>>>(uRe, uIm, vRe, vIm, out);
}